// RelativePositionEncoder_33706903339527
// MI455X (gfx1250) — compile-verified
//
#include <hip/hip_runtime.h>
#include <stdint.h>

// Problem constants (from reference)
#define NN       1024
#define ZD       128
#define ZV       32          // float4 per 128-float row
#define R_MAX    32
#define S_MAX    2
#define NUM_POS  66          // 2*R_MAX+2
#define NUM_TOK  66
// W rows: [0,66) = W_pos, [66,132) = W_tok, 132 = w_ent, [133,139) = W_chain

// LDS-staged W rows 66..138 (tok + ent + chain) = 73 rows
#define STAGE_ROWS 73
#define STAGE_F4   (STAGE_ROWS * ZV)   // 2336 vec4 = 37,376 B

typedef float v4f __attribute__((ext_vector_type(4)));

__global__ __launch_bounds__(256)
void relpos_encode_kernel(const int* __restrict__ asym,
                          const int* __restrict__ resi,
                          const int* __restrict__ ent,
                          const int* __restrict__ tok,
                          const int* __restrict__ sym,
                          const float* __restrict__ W,
                          float* __restrict__ out)
{
    __shared__ v4f sWr[STAGE_F4];               // W rows 66..138
    __shared__ int sA[NN], sR[NN], sE[NN], sT[NN], sS[NN];

    const int tid = threadIdx.x;

    // ---- async stage: W rows [66,139) as b128 straight into LDS ----
    {
        const v4f* src = (const v4f*)W + NUM_POS * ZV;
        const uint32_t lb = (uint32_t)(uintptr_t)&sWr[0];
        for (int k = tid; k < STAGE_F4; k += 256) {
            uint64_t g = (uint64_t)(uintptr_t)(src + k);
            uint32_t l = lb + 16u * (uint32_t)k;
            asm volatile("global_load_async_to_lds_b128 %0, %1, off"
                         :: "v"(l), "v"(g) : "memory");
        }
    }
    // ---- async stage: the five id arrays (b32) ----
#define STAGE_IDS(SRC, DST)                                                  \
    {                                                                        \
        const uint32_t lb = (uint32_t)(uintptr_t)&DST[0];                    \
        for (int k = tid; k < NN; k += 256) {                                \
            uint64_t g = (uint64_t)(uintptr_t)(SRC + k);                     \
            uint32_t l = lb + 4u * (uint32_t)k;                              \
            asm volatile("global_load_async_to_lds_b32 %0, %1, off"          \
                         :: "v"(l), "v"(g) : "memory");                      \
        }                                                                    \
    }
    STAGE_IDS(asym, sA)
    STAGE_IDS(resi, sR)
    STAGE_IDS(ent,  sE)
    STAGE_IDS(tok,  sT)
    STAGE_IDS(sym,  sS)
#undef STAGE_IDS

    asm volatile("s_wait_asynccnt 0" ::: "memory");
    __syncthreads();

    // ---- one block per row i; 8 waves; each wave owns pair (i,j) ----
    const int i    = blockIdx.x;
    const int lane = tid & 31;
    const int wave = tid >> 5;

    const int a_i = sA[i], r_i = sR[i], e_i = sE[i], t_i = sT[i], s_i = sS[i];

    const v4f* Wpos = (const v4f*)W;                     // rows 0..65 via cache
    const v4f  went = sWr[NUM_TOK * ZV + lane];          // w_ent row (LDS)

    v4f* orow = (v4f*)out + (size_t)i * (NN * ZV);

#pragma unroll 2
    for (int j = wave; j < NN; j += 8) {
        const int a_j = sA[j], r_j = sR[j], e_j = sE[j], t_j = sT[j], s_j = sS[j];

        const bool sc = (a_i == a_j);
        int dres = min(max(r_i - r_j + R_MAX, 0), 2 * R_MAX);
        if (!sc) dres = 2 * R_MAX + 1;
        int dtok = min(max(t_i - t_j + R_MAX, 0), 2 * R_MAX);
        if (!(sc && (r_i == r_j))) dtok = 2 * R_MAX + 1;
        int dch = min(max(s_i - s_j + S_MAX, 0), 2 * S_MAX);
        if (sc) dch = 2 * S_MAX + 1;
        const float ef = (e_i == e_j) ? 1.0f : 0.0f;

        const v4f p  = Wpos[dres * ZV + lane];                     // global (hot)
        const v4f tt = sWr[dtok * ZV + lane];                      // LDS
        const v4f cc = sWr[(NUM_TOK + 1 + dch) * ZV + lane];       // LDS

        const v4f r = p + tt + cc + ef * went;

        // streaming 512 MB output: non-temporal store, keep L2 for W
        __builtin_nontemporal_store(r, &orow[(size_t)j * ZV + lane]);
    }
}

extern "C" void kernel_launch(void* const* d_in, const int* in_sizes, int n_in,
                              void* d_out, int out_size, void* d_ws, size_t ws_size,
                              hipStream_t stream) {
    const int*   asym = (const int*)d_in[0];
    const int*   resi = (const int*)d_in[1];
    const int*   ent  = (const int*)d_in[2];
    const int*   tok  = (const int*)d_in[3];
    const int*   sym  = (const int*)d_in[4];
    const float* W    = (const float*)d_in[5];
    float*       out  = (float*)d_out;

    relpos_encode_kernel<<<dim3(NN), dim3(256), 0, stream>>>(
        asym, resi, ent, tok, sym, W, out);
}